// Denoiser_59090160058537
// MI455X (gfx1250) — compile-verified
//
#include <hip/hip_runtime.h>
#include <math.h>

// Problem constants (from the reference)
#define NN   4096
#define DD   256
#define KK   32
#define LL   4
#define TT   100
#define PP   64
#define BB   32
#define ND   ((size_t)NN * DD)

typedef __attribute__((ext_vector_type(2))) float v2f;
typedef __attribute__((ext_vector_type(8))) float v8f;

// ---------------------------------------------------------------------------
// Kernel 1: tvec[n] = sum_t basis(t) * W_in[t, n]   (time embedding folded in)
// ---------------------------------------------------------------------------
__global__ void tvec_kernel(const float* __restrict__ tptr,
                            const float* __restrict__ W_in,
                            float* __restrict__ tvec) {
    float tval = tptr[0];
    int n = threadIdx.x; // 0..255
    float acc = 0.0f;
    for (int tt = 0; tt < TT; ++tt) {
        float d = tval * 99.0f - (float)tt;          // (t - c)/step, step = 1/99
        float b = 0.0f;
        if (fabsf(d) < 1.0f) {
            float c = cosf(1.5707963267948966f * d); // cos(pi*d/2)
            b = c * c;
        }
        acc += b * W_in[(size_t)tt * DD + n];
    }
    tvec[n] = acc;
}

// ---------------------------------------------------------------------------
// Kernel 2: kNN (one block per node) + gaussian RBF
// ---------------------------------------------------------------------------
__global__ __launch_bounds__(256) void knn_kernel(const float* __restrict__ coord,
                                                  int* __restrict__ idxp,
                                                  float* __restrict__ rbf) {
    __shared__ float d2s[NN];
    __shared__ float rmin[256];
    __shared__ int   ridx[256];
    __shared__ float bdist[KK];
    int i = blockIdx.x;
    int tid = threadIdx.x;
    float cx = coord[i * 3 + 0];
    float cy = coord[i * 3 + 1];
    float cz = coord[i * 3 + 2];
    for (int j = tid; j < NN; j += 256) {
        float dx = cx - coord[j * 3 + 0];
        float dy = cy - coord[j * 3 + 1];
        float dz = cz - coord[j * 3 + 2];
        float dd = dx * dx + dy * dy + dz * dz;
        d2s[j] = (j == i) ? 1.0e9f : dd;
    }
    __syncthreads();
    for (int k = 0; k < KK; ++k) {
        float bm = 3.0e38f;
        int   bi = NN;
        for (int j = tid; j < NN; j += 256) {
            float v = d2s[j];
            if (v < bm) { bm = v; bi = j; }
        }
        rmin[tid] = bm; ridx[tid] = bi;
        __syncthreads();
        for (int s = 128; s > 0; s >>= 1) {
            if (tid < s) {
                float ov = rmin[tid + s]; int oi = ridx[tid + s];
                if (ov < rmin[tid] || (ov == rmin[tid] && oi < ridx[tid])) {
                    rmin[tid] = ov; ridx[tid] = oi;
                }
            }
            __syncthreads();
        }
        if (tid == 0) {
            int b = ridx[0];
            idxp[i * KK + k] = b;
            bdist[k] = sqrtf(fmaxf(rmin[0], 1.0e-12f));
            d2s[b] = 3.0e38f;
        }
        __syncthreads();
    }
    // rbf[i,k,b] = exp(-((d - 32*b/31)/1)^2)
    for (int q = tid; q < KK * BB; q += 256) {
        int k = q / BB, b = q % BB;
        float c = 32.0f * (float)b / 31.0f;
        float dd = bdist[k] - c;
        rbf[(size_t)i * KK * BB + q] = expf(-dd * dd);
    }
}

// ---------------------------------------------------------------------------
// Kernel 3: generic fp32 WMMA GEMM, one 16x16 output tile per wave.
//   C[M,Nc] = A[M,K]@B[K,Nc] (+ A2[M,K2]@B2[K2,Nc]) (+ rowScale[m]*rowAdd[n])
//   (+ C if accumulate). K stepped by 4 via V_WMMA_F32_16X16X4_F32.
// A layout (ISA 7.12.2): lane m = lane&15, holds K = kk + 2*(lane>>4) + {0,1}
// B layout (symmetric):  lane n = lane&15, holds K = kk + 2*(lane>>4) + {0,1}
// C/D: 8 VGPRs; lanes 0-15 rows 0-7, lanes 16-31 rows 8-15; col = lane&15.
// ---------------------------------------------------------------------------
__global__ __launch_bounds__(128) void wmma_gemm(
    const float* __restrict__ A, int lda,
    const float* __restrict__ B, int ldb, int K,
    const float* __restrict__ A2, int lda2,
    const float* __restrict__ B2, int ldb2, int K2,
    float* __restrict__ C, int ldc, int tilesN,
    const float* __restrict__ rowAdd,
    const float* __restrict__ rowScale,
    int accumulate) {
    int wid  = threadIdx.x >> 5;
    int lane = threadIdx.x & 31;
    int tile = blockIdx.x * 4 + wid;
    int tm = tile / tilesN;
    int tn = tile % tilesN;
    int m0 = tm * 16, n0 = tn * 16;
    int mr   = lane & 15;
    int koff = (lane >> 4) << 1; // 0 or 2

    v8f acc = {};
    for (int kk = 0; kk < K; kk += 4) {
        v2f a, b;
        const float* ap = A + (size_t)(m0 + mr) * lda + kk + koff;
        a.x = ap[0];
        a.y = ap[1];
        b.x = B[(size_t)(kk + koff)     * ldb + n0 + mr];
        b.y = B[(size_t)(kk + koff + 1) * ldb + n0 + mr];
        acc = __builtin_amdgcn_wmma_f32_16x16x4_f32(false, a, false, b,
                                                    (short)0, acc, false, false);
    }
    if (A2) {
        for (int kk = 0; kk < K2; kk += 4) {
            v2f a, b;
            const float* ap = A2 + (size_t)(m0 + mr) * lda2 + kk + koff;
            a.x = ap[0];
            a.y = ap[1];
            b.x = B2[(size_t)(kk + koff)     * ldb2 + n0 + mr];
            b.y = B2[(size_t)(kk + koff + 1) * ldb2 + n0 + mr];
            acc = __builtin_amdgcn_wmma_f32_16x16x4_f32(false, a, false, b,
                                                        (short)0, acc, false, false);
        }
    }
    int rbase = m0 + ((lane >> 4) << 3);
    int col   = n0 + mr;
    for (int r = 0; r < 8; ++r) {
        int row = rbase + r;
        float v = acc[r];
        if (rowAdd) v += (rowScale ? rowScale[row] : 1.0f) * rowAdd[col];
        size_t o = (size_t)row * ldc + col;
        if (accumulate) v += C[o];
        C[o] = v;
    }
}

// ---------------------------------------------------------------------------
// Kernel 4: row LayerNorm (D=256), wave-per-row, optional mask multiply
// ---------------------------------------------------------------------------
__global__ __launch_bounds__(256) void ln_kernel(const float* __restrict__ X,
                                                 float* __restrict__ Y,
                                                 const float* __restrict__ mask) {
    int wid  = threadIdx.x >> 5;
    int lane = threadIdx.x & 31;
    int row  = blockIdx.x * 8 + wid;
    const float* x = X + (size_t)row * DD;
    float v[8];
    float s = 0.0f;
    #pragma unroll
    for (int j = 0; j < 8; ++j) { v[j] = x[lane + j * 32]; s += v[j]; }
    #pragma unroll
    for (int o = 16; o > 0; o >>= 1) s += __shfl_xor(s, o, 32);
    float mean = s * (1.0f / 256.0f);
    float q = 0.0f;
    #pragma unroll
    for (int j = 0; j < 8; ++j) { float d = v[j] - mean; q += d * d; }
    #pragma unroll
    for (int o = 16; o > 0; o >>= 1) q += __shfl_xor(q, o, 32);
    float rinv = rsqrtf(q * (1.0f / 256.0f) + 1.0e-5f);
    float mk = mask ? mask[row] : 1.0f;
    float* y = Y + (size_t)row * DD;
    #pragma unroll
    for (int j = 0; j < 8; ++j) y[lane + j * 32] = (v[j] - mean) * rinv * mk;
}

// ---------------------------------------------------------------------------
// Kernel 5: fused edge-weight + gather + mean:
//   agg_pre[i,d] = (1/K) * sum_k hs[idx[i,k], d] * (rbf[i,k,:] @ W_rad[:,d])
// One block per node; W_rad (32x256) and rbf (32x32) staged in LDS.
// ---------------------------------------------------------------------------
__global__ __launch_bounds__(256) void conv_gather(
    const float* __restrict__ hs, const int* __restrict__ idxp,
    const float* __restrict__ rbf, const float* __restrict__ Wrad,
    float* __restrict__ agg_pre) {
    __shared__ float rs[KK * BB];
    __shared__ float ws[BB * DD];
    __shared__ int   is[KK];
    int i = blockIdx.x, tid = threadIdx.x;
    for (int q = tid; q < KK * BB; q += 256) rs[q] = rbf[(size_t)i * KK * BB + q];
    for (int q = tid; q < BB * DD; q += 256) ws[q] = Wrad[q];
    if (tid < KK) is[tid] = idxp[i * KK + tid];
    __syncthreads();
    float acc = 0.0f;
    for (int k = 0; k < KK; ++k) {
        float ew = 0.0f;
        #pragma unroll
        for (int b = 0; b < BB; ++b) ew += rs[k * BB + b] * ws[b * DD + tid];
        acc += hs[(size_t)is[k] * DD + tid] * ew;
    }
    agg_pre[(size_t)i * DD + tid] = acc * (1.0f / (float)KK);
}

// ---------------------------------------------------------------------------
// Kernel 6: coord head (256 -> 3), written into d_out cols 256..258
// ---------------------------------------------------------------------------
__global__ void coord_kernel(const float* __restrict__ outbuf,
                             const float* __restrict__ W_coord,
                             float* __restrict__ d_out) {
    int i = blockIdx.x * blockDim.x + threadIdx.x;
    if (i >= NN) return;
    float a0 = 0.0f, a1 = 0.0f, a2 = 0.0f;
    const float* o = outbuf + (size_t)i * DD;
    for (int d = 0; d < DD; ++d) {
        float v = o[d];
        a0 += v * W_coord[d * 3 + 0];
        a1 += v * W_coord[d * 3 + 1];
        a2 += v * W_coord[d * 3 + 2];
    }
    float* dst = d_out + (size_t)i * (DD + 3) + DD;
    dst[0] = a0; dst[1] = a1; dst[2] = a2;
}

// ---------------------------------------------------------------------------
extern "C" void kernel_launch(void* const* d_in, const int* in_sizes, int n_in,
                              void* d_out, int out_size, void* d_ws, size_t ws_size,
                              hipStream_t stream) {
    (void)in_sizes; (void)n_in; (void)out_size; (void)ws_size;
    const float* feat    = (const float*)d_in[0];
    const float* coord   = (const float*)d_in[1];
    const float* mask    = (const float*)d_in[2];
    const float* tptr    = (const float*)d_in[3];
    const float* pos     = (const float*)d_in[4];
    const float* W_in    = (const float*)d_in[5];
    const float* W_self  = (const float*)d_in[6];
    const float* W_rad   = (const float*)d_in[7];
    const float* W_conv  = (const float*)d_in[8];
    const float* W_cat   = (const float*)d_in[9];
    const float* W_out   = (const float*)d_in[10];
    const float* W_noise = (const float*)d_in[11];
    const float* W_coord = (const float*)d_in[12];
    float* out = (float*)d_out;

    float* w       = (float*)d_ws;
    float* tvec    = w;            w += 256;
    float* states  = w;            w += 5 * ND;   // h0..h4, [N,D] each
    float* hs      = w;            w += ND;
    float* agg_pre = w;            w += ND;
    float* agg     = w;            w += ND;
    float* tmp     = w;            w += ND;
    float* outbuf  = w;            w += ND;
    float* rbf     = w;            w += (size_t)NN * KK * BB;
    int*   idx     = (int*)w;

    // tiles: M=4096 -> 256 row tiles; Nc=256 -> 16 col tiles; 4 waves/block
    const int tilesN = DD / 16;
    const int blocks = (NN / 16) * tilesN / 4; // 1024

    // 1) time embedding vector
    tvec_kernel<<<1, 256, 0, stream>>>(tptr, W_in, tvec);
    // 2) kNN graph + RBF
    knn_kernel<<<NN, 256, 0, stream>>>(coord, idx, rbf);
    // 3) h0 = LN(feat@W_in[100:356] + pos@W_in[356:420] + mask*tvec) * mask
    wmma_gemm<<<blocks, 128, 0, stream>>>(
        feat, DD, W_in + (size_t)TT * DD, DD, DD,
        pos, PP, W_in + (size_t)(TT + DD) * DD, DD, PP,
        tmp, DD, tilesN, tvec, mask, 0);
    ln_kernel<<<NN / 8, 256, 0, stream>>>(tmp, states, mask);

    for (int l = 0; l < LL; ++l) {
        const float* h       = states + (size_t)l * ND;
        float*       h_next  = states + (size_t)(l + 1) * ND;
        const float* Wself_l = W_self + (size_t)l * DD * DD;
        const float* Wrad_l  = W_rad  + (size_t)l * BB * DD;
        const float* Wconv_l = W_conv + (size_t)l * DD * DD;
        const float* Wcat_l  = W_cat  + (size_t)l * 2 * DD * DD;

        // hs = LN(h @ W_self[l])
        wmma_gemm<<<blocks, 128, 0, stream>>>(
            h, DD, Wself_l, DD, DD,
            nullptr, 0, nullptr, 0, 0,
            tmp, DD, tilesN, nullptr, nullptr, 0);
        ln_kernel<<<NN / 8, 256, 0, stream>>>(tmp, hs, nullptr);

        // agg_pre = mean_k hs[idx]*(rbf@W_rad)
        conv_gather<<<NN, 256, 0, stream>>>(hs, idx, rbf, Wrad_l, agg_pre);

        // agg = agg_pre @ W_conv[l]
        wmma_gemm<<<blocks, 128, 0, stream>>>(
            agg_pre, DD, Wconv_l, DD, DD,
            nullptr, 0, nullptr, 0, 0,
            agg, DD, tilesN, nullptr, nullptr, 0);

        // h_next = LN([h, agg] @ W_cat[l]) * mask
        wmma_gemm<<<blocks, 128, 0, stream>>>(
            h, DD, Wcat_l, DD, DD,
            agg, DD, Wcat_l + (size_t)DD * DD, DD, DD,
            tmp, DD, tilesN, nullptr, nullptr, 0);
        ln_kernel<<<NN / 8, 256, 0, stream>>>(tmp, h_next, mask);
    }

    // out = concat(states) @ W_out  == sum_l states_l @ W_out[l*256:...]
    for (int l = 0; l < LL + 1; ++l) {
        wmma_gemm<<<blocks, 128, 0, stream>>>(
            states + (size_t)l * ND, DD, W_out + (size_t)l * DD * DD, DD, DD,
            nullptr, 0, nullptr, 0, 0,
            outbuf, DD, tilesN, nullptr, nullptr, l == 0 ? 0 : 1);
    }

    // feat_out -> d_out[:, 0:256] (ldc = 259)
    wmma_gemm<<<blocks, 128, 0, stream>>>(
        outbuf, DD, W_noise, DD, DD,
        nullptr, 0, nullptr, 0, 0,
        out, DD + 3, tilesN, nullptr, nullptr, 0);

    // coord_out -> d_out[:, 256:259]
    coord_kernel<<<NN / 256, 256, 0, stream>>>(outbuf, W_coord, out);
}